// PoolingModule_45681272160839
// MI455X (gfx1250) — compile-verified
//
#include <hip/hip_runtime.h>
#include <stdint.h>

// Chunked mean pooling, g=4, DIM=1024, fp32.
// Streaming-bandwidth kernel: 640 MB total traffic, ~27.5 us floor at 23.3 TB/s.
// Data path: global -> LDS via CDNA5 async copies (ASYNCcnt), double buffered,
// non-temporal in/out (single-touch stream, keep GL2 clean).

typedef float v4f __attribute__((ext_vector_type(4)));

#define G           4
#define DIM         1024
#define BLOCK       256                 // 8 waves (wave32); DIM/4 float4 lanes per row
#define CPB         8                   // chunks per block (n_chunks divisible by CPB)
#define CHUNK_BYTES (G * DIM * 4)       // 16384 B, contiguous in memory
#define CHUNK_FLTS  (G * DIM)           // 4096 floats

__device__ __forceinline__ void async_copy_chunk(unsigned lds, const char* src)
{
    // 4 x b128 per thread x 256 threads = 16 KB. The instruction offset applies
    // to BOTH the global and the LDS address, so one (vdst, vaddr) pair covers
    // the whole contiguous copy. NT: input is touched exactly once.
    asm volatile(
        "global_load_async_to_lds_b128 %0, %1, off th:TH_LOAD_NT\n\t"
        "global_load_async_to_lds_b128 %0, %1, off offset:4096 th:TH_LOAD_NT\n\t"
        "global_load_async_to_lds_b128 %0, %1, off offset:8192 th:TH_LOAD_NT\n\t"
        "global_load_async_to_lds_b128 %0, %1, off offset:12288 th:TH_LOAD_NT"
        :: "v"(lds), "v"(src) : "memory");
}

__global__ __launch_bounds__(BLOCK) void mean_pool4_async(
    const float* __restrict__ x, float* __restrict__ out)
{
    __shared__ float smem[2 * CHUNK_FLTS];   // 32 KB double buffer

    const int tid  = threadIdx.x;
    const int base = blockIdx.x * CPB;

    // Numeric LDS byte address of smem: low 32 bits of the flat pointer
    // (ISA 10.2, LDS aperture).
    const unsigned lds0 = (unsigned)(uint64_t)(void*)smem + (unsigned)(tid * 16);
    const char*    g0   = (const char*)x + (size_t)base * CHUNK_BYTES + tid * 16;

    // Prime the pipeline: chunk `base` -> buffer 0.
    async_copy_chunk(lds0, g0);

#pragma unroll
    for (int i = 0; i < CPB; ++i) {
        const unsigned cur = (unsigned)(i & 1);

        if (i + 1 < CPB) {
            // Issue next chunk into the other buffer, then drain only the
            // current chunk's 4 async ops (keep the new 4 in flight).
            async_copy_chunk(lds0 + (cur ^ 1u) * (unsigned)CHUNK_BYTES,
                             g0 + (size_t)(i + 1) * CHUNK_BYTES);
            asm volatile("s_wait_asynccnt 0x4" ::: "memory");
        } else {
            asm volatile("s_wait_asynccnt 0x0" ::: "memory");
        }
        __syncthreads();   // all waves' async data visible in LDS

        // thread t owns output floats [4t, 4t+4) of this chunk's output row
        const v4f* s   = (const v4f*)(smem + cur * CHUNK_FLTS);
        v4f        acc = s[tid] + s[tid + 256] + s[tid + 512] + s[tid + 768];
        acc *= 0.25f;

        // streaming output: non-temporal 128-bit store
        __builtin_nontemporal_store(acc, (v4f*)(out + (size_t)(base + i) * DIM) + tid);

        if (i + 1 < CPB)
            __syncthreads();   // all reads of buf `cur` done before it is refilled
    }
}

extern "C" void kernel_launch(void* const* d_in, const int* in_sizes, int n_in,
                              void* d_out, int out_size, void* d_ws, size_t ws_size,
                              hipStream_t stream)
{
    const float* x   = (const float*)d_in[0];
    float*       out = (float*)d_out;

    const int total_rows = in_sizes[0] / DIM;      // 131072
    const int n_chunks   = total_rows / G;         // 32768 output rows
    const int grid       = n_chunks / CPB;         // 4096 blocks (exact)

    mean_pool4_async<<<grid, BLOCK, 0, stream>>>(x, out);
}